// GraphSageLayer_6605659701688
// MI455X (gfx1250) — compile-verified
//
#include <hip/hip_runtime.h>

typedef __attribute__((ext_vector_type(2))) float v2f;
typedef __attribute__((ext_vector_type(8))) float v8f;

#define NNODES 100000
#define NEDGES 1600000
#define DIM    128

// ---------------------------------------------------------------------------
// Kernel 1: zero the workspace accumulators (deg[N], neigh[N*D], esum[N*D])
// ---------------------------------------------------------------------------
__global__ void gsage_zero_ws(float4* __restrict__ p, int n4) {
    int i = blockIdx.x * blockDim.x + threadIdx.x;
    if (i < n4) p[i] = make_float4(0.f, 0.f, 0.f, 0.f);
}

// ---------------------------------------------------------------------------
// Kernel 2: edge scatter. 32 lanes per edge, each lane owns a float4 chunk of
// the 128-wide feature. Accumulates:
//   neigh[dst] += nfeat[src] * w      (weighted neighbor message)
//   esum [dst] += efeat[e]            (edge-feature segment sum)
//   deg  [dst] += 1                   (in-degree, lane 0 only)
// ---------------------------------------------------------------------------
__global__ void gsage_scatter(const float* __restrict__ nfeat,
                              const float* __restrict__ efeat,
                              const float* __restrict__ ew,
                              const int*   __restrict__ src,
                              const int*   __restrict__ dst,
                              float* __restrict__ deg,
                              float* __restrict__ neigh,
                              float* __restrict__ esum) {
    unsigned tid = blockIdx.x * blockDim.x + threadIdx.x;
    unsigned e = tid >> 5;
    unsigned c = tid & 31u;
    if (e >= NEDGES) return;

    const int   s = src[e];
    const int   d = dst[e];
    const float w = ew[e];
    const int   k = (int)c * 4;

    float4 nv = *(const float4*)(nfeat + (size_t)s * DIM + k);
    float4 ev = *(const float4*)(efeat + (size_t)e * DIM + k);

    float* np = neigh + (size_t)d * DIM + k;
    atomicAdd(np + 0, nv.x * w);
    atomicAdd(np + 1, nv.y * w);
    atomicAdd(np + 2, nv.z * w);
    atomicAdd(np + 3, nv.w * w);

    float* ep = esum + (size_t)d * DIM + k;
    atomicAdd(ep + 0, ev.x);
    atomicAdd(ep + 1, ev.y);
    atomicAdd(ep + 2, ev.z);
    atomicAdd(ep + 3, ev.w);

    if (c == 0) atomicAdd(deg + d, 1.0f);
}

// ---------------------------------------------------------------------------
// Kernel 3: fused double-GEMM + epilogue on the matrix pipe.
//   out = ((neigh + 2*nfeat)/(deg+1)) @ Wn^T + b_neigh
//       + ( esum / max(deg,1)       ) @ We^T + (deg>0 ? b_edge : 0)
// One wave = one 16x16 output tile. 8 waves/block cover all 8 column tiles of
// a 16-row node tile. Both GEMMs accumulate into one v8f via
// v_wmma_f32_16x16x4_f32 (fp32-exact, K=4 per step, 64 WMMAs per wave).
// ---------------------------------------------------------------------------
__global__ void __launch_bounds__(256) gsage_gemm(
    const float* __restrict__ nfeat,
    const float* __restrict__ Wn,
    const float* __restrict__ bn,
    const float* __restrict__ We,
    const float* __restrict__ be,
    const float* __restrict__ deg,
    const float* __restrict__ neigh,
    const float* __restrict__ esum,
    float* __restrict__ out) {

    const int wave = threadIdx.x >> 5;      // 0..7 -> output column tile
    const int lane = threadIdx.x & 31;
    const int j0   = wave * 16;
    const int m0   = blockIdx.x * 16;

    const int lane16 = lane & 15;
    const int khalf  = (lane >> 4) * 2;     // A/B frag: lanes 0-15 K-low pair,
                                            //           lanes 16-31 K-high pair
    const int arow = m0 + lane16;           // A-fragment source row (M = lane%16)
    const int bcol = j0 + lane16;           // B-fragment source col (N = lane%16)

    const float dg   = deg[arow];
    const float inv1 = 1.0f / (dg + 1.0f);
    const float inv2 = 1.0f / fmaxf(dg, 1.0f);

    const float* nrow = nfeat + (size_t)arow * DIM;
    const float* grow = neigh + (size_t)arow * DIM;
    const float* erow = esum  + (size_t)arow * DIM;
    const float* wnr  = Wn    + (size_t)bcol * DIM;  // B[k][j] = Wn[j*D + k]
    const float* wer  = We    + (size_t)bcol * DIM;

    v8f acc = {0.f, 0.f, 0.f, 0.f, 0.f, 0.f, 0.f, 0.f};

    #pragma unroll 4
    for (int k0 = 0; k0 < DIM; k0 += 4) {
        const int k = k0 + khalf;

        // GEMM 1: x1 = (neigh + 2*nfeat) * inv1, weights Wn^T
        float2 gg = *(const float2*)(grow + k);
        float2 nn = *(const float2*)(nrow + k);
        float2 w1 = *(const float2*)(wnr + k);
        v2f a1, b1;
        a1.x = (gg.x + 2.0f * nn.x) * inv1;
        a1.y = (gg.y + 2.0f * nn.y) * inv1;
        b1.x = w1.x; b1.y = w1.y;
        acc = __builtin_amdgcn_wmma_f32_16x16x4_f32(
                  false, a1, false, b1, (short)0, acc, false, false);

        // GEMM 2: x2 = esum * inv2, weights We^T
        float2 ee = *(const float2*)(erow + k);
        float2 w2 = *(const float2*)(wer + k);
        v2f a2, b2;
        a2.x = ee.x * inv2;
        a2.y = ee.y * inv2;
        b2.x = w2.x; b2.y = w2.y;
        acc = __builtin_amdgcn_wmma_f32_16x16x4_f32(
                  false, a2, false, b2, (short)0, acc, false, false);
    }

    // Epilogue: C/D layout — VGPR r, lanes 0-15 -> M=r, lanes 16-31 -> M=8+r;
    // N = lane%16. Add b_neigh always, b_edge only where in-degree > 0.
    const float bno  = bn[bcol];
    const float beo  = be[bcol];
    const int   half = (lane >> 4) * 8;

    #pragma unroll
    for (int r = 0; r < 8; ++r) {
        const int   row = m0 + half + r;
        const float dgr = deg[row];
        const float v   = acc[r] + bno + ((dgr > 0.0f) ? beo : 0.0f);
        out[(size_t)row * DIM + bcol] = v;
    }
}

// ---------------------------------------------------------------------------
extern "C" void kernel_launch(void* const* d_in, const int* in_sizes, int n_in,
                              void* d_out, int out_size, void* d_ws, size_t ws_size,
                              hipStream_t stream) {
    const float* nfeat = (const float*)d_in[0];
    const float* efeat = (const float*)d_in[1];
    const float* ew    = (const float*)d_in[2];
    const float* Wn    = (const float*)d_in[3];
    const float* bn    = (const float*)d_in[4];
    const float* We    = (const float*)d_in[5];
    const float* be    = (const float*)d_in[6];
    const int*   src   = (const int*)d_in[7];
    const int*   dst   = (const int*)d_in[8];
    float*       out   = (float*)d_out;

    // Workspace layout: deg[N] | neigh[N*D] | esum[N*D]  (~103 MB)
    float* ws    = (float*)d_ws;
    float* deg   = ws;
    float* neigh = ws + NNODES;
    float* esum  = neigh + (size_t)NNODES * DIM;

    // 1) zero accumulators
    const int n4 = (NNODES * (1 + 2 * DIM)) / 4;   // 25.7M floats / 4
    gsage_zero_ws<<<(n4 + 255) / 256, 256, 0, stream>>>((float4*)d_ws, n4);

    // 2) edge scatter: 32 lanes per edge
    const unsigned sthreads = (unsigned)NEDGES * 32u;   // 51.2M
    gsage_scatter<<<sthreads / 256, 256, 0, stream>>>(
        nfeat, efeat, ew, src, dst, deg, neigh, esum);

    // 3) fused WMMA double-GEMM + epilogue: one block per 16-node tile
    gsage_gemm<<<NNODES / 16, 256, 0, stream>>>(
        nfeat, Wn, bn, We, be, deg, neigh, esum, out);
}